// STSourceModule_65618510348871
// MI455X (gfx1250) — compile-verified
//
#include <hip/hip_runtime.h>
#include <hip/hip_bf16.h>
#include <math.h>

// Problem constants (match reference setup_inputs)
#define B_    8
#define S_    256
#define N_    128
#define C_    64
#define H_    4
#define AH_   128
#define HID_  256
#define FH_   32
#define MAX_SP_ 180.0f
#define MAX_TP_ 365.0f

typedef float v2f __attribute__((ext_vector_type(2)));
typedef float v8f __attribute__((ext_vector_type(8)));

__device__ __forceinline__ float silu_f(float z) { return z / (1.0f + expf(-z)); }

// ---------------------------------------------------------------------------
// Kernel 1: per (b,n) FIRE bias -> e_n = exp(bias); per (b,c) Etot and mask.
// ---------------------------------------------------------------------------
__global__ __launch_bounds__(128) void k_setup(
    const int* __restrict__ labels,
    const float* __restrict__ dsp, const float* __restrict__ dtp,
    const float* __restrict__ c_sp,
    const float* __restrict__ sp_w1, const float* __restrict__ sp_w2,
    const float* __restrict__ c_tp,
    const float* __restrict__ tp_w1, const float* __restrict__ tp_w2,
    float* __restrict__ e_out, float* __restrict__ etot_out,
    float* __restrict__ mask_out)
{
    int b = blockIdx.x;
    int n = threadIdx.x;
    __shared__ float sE[C_];
    __shared__ int   sCnt[C_];
    if (n < C_) { sE[n] = 0.0f; sCnt[n] = 0; }

    // FIRE spatial
    float csp = fmaxf(c_sp[0], 0.0f);
    float d1  = dsp[b * N_ + n];
    float dn1 = logf(csp * d1 + 1.0f) / logf(csp * MAX_SP_ + 1.0f);
    float acc1 = 0.0f;
    for (int i = 0; i < FH_; ++i) acc1 += silu_f(dn1 * sp_w1[i]) * sp_w2[i];
    // FIRE temporal
    float ctp = fmaxf(c_tp[0], 0.0f);
    float d2  = dtp[b * N_ + n];
    float dn2 = logf(ctp * d2 + 1.0f) / logf(ctp * MAX_TP_ + 1.0f);
    float acc2 = 0.0f;
    for (int i = 0; i < FH_; ++i) acc2 += silu_f(dn2 * tp_w1[i]) * tp_w2[i];

    float ev = expf(acc1 + acc2);
    e_out[b * N_ + n] = ev;
    __syncthreads();

    int lab = labels[b * N_ + n];
    atomicAdd(&sE[lab], ev);
    atomicAdd(&sCnt[lab], 1);
    __syncthreads();

    if (n < C_) {
        etot_out[b * C_ + n] = sE[n];
        mask_out[b * C_ + n] = (sCnt[n] > 0) ? 1.0f : 0.0f;
    }
}

// ---------------------------------------------------------------------------
// Kernel 2: fold value/ffn weights:
//   base[j] = ffn_b[j] + sum_k ffn_w[j,k]*val_b[k]
//   W[h][j] = sum_d ffn_w[j, h*32+d]*val_w[h*32+d]
// ---------------------------------------------------------------------------
__global__ __launch_bounds__(256) void k_fold(
    const float* __restrict__ ffn_w, const float* __restrict__ ffn_b,
    const float* __restrict__ val_w, const float* __restrict__ val_b,
    float* __restrict__ base_out, float* __restrict__ W_out)
{
    int j = threadIdx.x;
    const float* row = ffn_w + (size_t)j * AH_;
    float b0 = ffn_b[j];
    float wh[H_] = {0.f, 0.f, 0.f, 0.f};
    for (int k = 0; k < AH_; ++k) {
        float w = row[k];
        b0        += w * val_b[k];
        wh[k >> 5] += w * val_w[k];
    }
    base_out[j] = b0;
    for (int h = 0; h < H_; ++h) W_out[h * HID_ + j] = wh[h];
}

// ---------------------------------------------------------------------------
// Kernel 3: per (b,s): E1[c] via LDS atomics; q[c][h] = gh*E1/(gh*E1+E0);
// then out[c, 0:256] = mask*(base + q . W) as 16x16 tiles via
// V_WMMA_F32_16X16X4_F32 (A = 16x4 q block, B = 4x16 W slice, C = mask*base).
// 256 threads = 8 wave32s: wave = (row-group 0..3) x (column half 0..1).
// ---------------------------------------------------------------------------
__global__ __launch_bounds__(256) void k_main(
    const int* __restrict__ src, const int* __restrict__ labels,
    const float* __restrict__ attn_w,
    const float* __restrict__ e_in, const float* __restrict__ etot_in,
    const float* __restrict__ mask_in,
    const float* __restrict__ base_in, const float* __restrict__ W_in,
    float* __restrict__ out)
{
    __shared__ float sE1[C_];
    __shared__ float sQ[C_ * H_];
    __shared__ float sM[C_];
    __shared__ float sBase[HID_];

    int tid = threadIdx.x;
    int bs  = blockIdx.x;        // flattened (b,s)
    int b   = bs / S_;

    if (tid < C_) { sE1[tid] = 0.0f; sM[tid] = mask_in[b * C_ + tid]; }
    sBase[tid] = base_in[tid];
    __syncthreads();

    // E1[c] = sum over member sites with x==1 of e_n
    if (tid < N_) {
        int x = src[(size_t)bs * N_ + tid];
        if (x) atomicAdd(&sE1[labels[b * N_ + tid]], e_in[b * N_ + tid]);
    }
    __syncthreads();

    // q[c][h], masked; 256 threads = 64 clusters x 4 heads exactly
    {
        int c = tid >> 2, h = tid & 3;
        float gh  = expf(attn_w[h]);
        float e1  = sE1[c];
        float et  = etot_in[b * C_ + c];
        float e0  = et - e1;
        float num = gh * e1;
        float den = num + e0;
        float q   = (den > 0.0f) ? (num / den) : 0.0f;
        sQ[tid] = q * sM[c];
    }
    __syncthreads();

    int lane = tid & 31;
    int wave = tid >> 5;
    int rg   = wave & 3;        // row group: clusters rg*16 .. rg*16+15
    int half = wave >> 1 >> 1;  // column half: tiles half*8 .. half*8+7
    int rowN = lane & 15;       // M (for A) / N (for B,C,D) within tile
    int hi   = lane >> 4;       // lane half selects K-pair {0,1} vs {2,3}

    // A (16x4 f32): lanes 0-15: VGPR0=K0, VGPR1=K1; lanes 16-31: K2, K3
    int c = rg * 16 + rowN;
    v2f a;
    a.x = sQ[c * 4 + 2 * hi];
    a.y = sQ[c * 4 + 2 * hi + 1];

    // Per-lane row masks for the C matrix (D rows m = v + 8*hi)
    float mrow[8];
#pragma unroll
    for (int v = 0; v < 8; ++v) mrow[v] = sM[rg * 16 + v + 8 * hi];

    float* orow = out + (size_t)bs * (C_ * HID_);

#pragma unroll
    for (int t = half * 8; t < half * 8 + 8; ++t) {
        int j = t * 16 + rowN;  // output column
        // B (4x16 f32): lanes 0-15: VGPR0=K0 row, VGPR1=K1; lanes 16-31: K2, K3
        v2f bb;
        bb.x = W_in[(2 * hi) * HID_ + j];
        bb.y = W_in[(2 * hi + 1) * HID_ + j];
        float bj = sBase[j];
        v8f cc;
#pragma unroll
        for (int v = 0; v < 8; ++v) cc[v] = mrow[v] * bj;

        v8f dd = __builtin_amdgcn_wmma_f32_16x16x4_f32(
            false, a, false, bb, (short)0, cc, false, false);

#pragma unroll
        for (int v = 0; v < 8; ++v) {
            int m = v + 8 * hi;
            orow[(size_t)(rg * 16 + m) * HID_ + j] = dd[v];
        }
    }
}

// ---------------------------------------------------------------------------
extern "C" void kernel_launch(void* const* d_in, const int* in_sizes, int n_in,
                              void* d_out, int out_size, void* d_ws, size_t ws_size,
                              hipStream_t stream)
{
    (void)in_sizes; (void)n_in; (void)out_size; (void)ws_size;

    const int*   src    = (const int*)  d_in[0];   // (B,S,N)
    const int*   labels = (const int*)  d_in[1];   // (B,N)
    const float* dsp    = (const float*)d_in[2];   // (B,N)
    const float* dtp    = (const float*)d_in[3];   // (B,N)
    // d_in[4] = num_clusters (compile-time 64 here)
    const float* c_sp   = (const float*)d_in[5];
    const float* sp_w1  = (const float*)d_in[6];   // (32,1)
    const float* sp_w2  = (const float*)d_in[7];   // (1,32)
    const float* c_tp   = (const float*)d_in[8];
    const float* tp_w1  = (const float*)d_in[9];
    const float* tp_w2  = (const float*)d_in[10];
    const float* attn_w = (const float*)d_in[11];  // (4,)
    // d_in[12] = attn_b : cancels inside the per-cluster softmax
    const float* val_w  = (const float*)d_in[13];  // (128,)
    const float* val_b  = (const float*)d_in[14];  // (128,)
    const float* ffn_w  = (const float*)d_in[15];  // (256,128)
    const float* ffn_b  = (const float*)d_in[16];  // (256,)

    float* out = (float*)d_out;

    // Workspace layout (floats): e | Etot | mask | base | W
    float* ws     = (float*)d_ws;
    float* e_buf  = ws;                       // B*N   = 1024
    float* etot   = e_buf + B_ * N_;          // B*C   = 512
    float* maskv  = etot  + B_ * C_;          // B*C   = 512
    float* basev  = maskv + B_ * C_;          // HID   = 256
    float* Wv     = basev + HID_;             // H*HID = 1024

    k_setup<<<B_, N_, 0, stream>>>(labels, dsp, dtp, c_sp, sp_w1, sp_w2,
                                   c_tp, tp_w1, tp_w2, e_buf, etot, maskv);
    k_fold<<<1, HID_, 0, stream>>>(ffn_w, ffn_b, val_w, val_b, basev, Wv);
    k_main<<<B_ * S_, 256, 0, stream>>>(src, labels, attn_w, e_buf, etot,
                                        maskv, basev, Wv, out);
}